// MPNN_QM9_35313221108120
// MI455X (gfx1250) — compile-verified
//
#include <hip/hip_runtime.h>

#define DEVINL __device__ __forceinline__

typedef float v8f __attribute__((ext_vector_type(8)));
typedef float v2f __attribute__((ext_vector_type(2)));

constexpr int Nn   = 50000;
constexpr int Ee   = 800000;
constexpr int Gg   = 2500;
constexpr int F_E  = 4;
constexpr int Hh   = 64;
constexpr int OUTd = 12;

// ---------------------------------------------------------------------------
// fp32 WMMA wrapper: D(16x16,f32) = A(16x4,f32) * B(4x16,f32) + C
// A frag: lane<16 -> {K=k, k+1} of row (lane&15); lane>=16 -> {K=k+2, k+3}
// B frag: lane<16 -> {K=k, k+1} of col (lane&15); lane>=16 -> {K=k+2, k+3}
// C/D   : VGPR r, lane<16 -> (M=r, N=lane); lane>=16 -> (M=r+8, N=lane-16)
// ---------------------------------------------------------------------------
DEVINL v8f wmma_f32(v2f a, v2f b, v8f c) {
  return __builtin_amdgcn_wmma_f32_16x16x4_f32(
      /*neg_a=*/false, a, /*neg_b=*/false, b,
      /*c_mod=*/(short)0, c, /*reuse_a=*/false, /*reuse_b=*/false);
}

// ---------------------------------------------------------------------------
// Zero-fill (workspace accumulators must be zeroed every call)
// ---------------------------------------------------------------------------
__global__ void zero_kernel(float* __restrict__ p, int n) {
  int i = blockIdx.x * blockDim.x + threadIdx.x;
  int stride = gridDim.x * blockDim.x;
  for (; i < n; i += stride) p[i] = 0.0f;
}

// ---------------------------------------------------------------------------
// A-fragment loaders. After full unrolling, k is a literal, so all region
// tests fold at compile time. Vector (b64) path only when row stride keeps
// 8-byte alignment (DIN % 4 == 0) and the 4-wide k-group doesn't straddle
// the concat boundary (always true for DIN=64 since boundaries are mult-4).
// ---------------------------------------------------------------------------
template <int DIN>
DEVINL float edge_a1(const float* __restrict__ x, const float* __restrict__ ea,
                     int s, int e, int idx) {
  if (idx < DIN) return x[s * DIN + idx];
  if (idx < DIN + F_E) return ea[e * F_E + (idx - DIN)];
  return 0.0f;
}

template <int DIN>
DEVINL v2f edge_a2(const float* __restrict__ x, const float* __restrict__ ea,
                   int s, int e, int k, int hh) {
  constexpr int K = DIN + F_E;
  constexpr bool VEC = (DIN % 4 == 0);
  if (VEC && (k + 3 < DIN))
    return *(const v2f*)&x[s * DIN + k + 2 * hh];
  if (VEC && (k >= DIN) && (k + 3 < K))
    return *(const v2f*)&ea[e * F_E + (k - DIN) + 2 * hh];
  v2f r;
  const int ia = k + 2 * hh;
  r.x = edge_a1<DIN>(x, ea, s, e, ia);
  r.y = edge_a1<DIN>(x, ea, s, e, ia + 1);
  return r;
}

template <int DIN>
DEVINL float upd_a1(const float* __restrict__ hin, const float* __restrict__ agg,
                    int n, int idx) {
  if (idx < DIN) return hin[n * DIN + idx];
  if (idx < DIN + Hh) return agg[n * Hh + (idx - DIN)];
  return 0.0f;
}

template <int DIN>
DEVINL v2f upd_a2(const float* __restrict__ hin, const float* __restrict__ agg,
                  int n, int k, int hh) {
  constexpr int K = DIN + Hh;
  constexpr bool VEC = (DIN % 4 == 0);
  if (VEC && (k + 3 < DIN))
    return *(const v2f*)&hin[n * DIN + k + 2 * hh];
  if (VEC && (k >= DIN) && (k + 3 < K))
    return *(const v2f*)&agg[n * Hh + (k - DIN) + 2 * hh];
  v2f r;
  const int ia = k + 2 * hh;
  r.x = upd_a1<DIN>(hin, agg, n, ia);
  r.y = upd_a1<DIN>(hin, agg, n, ia + 1);
  return r;
}

// ---------------------------------------------------------------------------
// Edge message + scatter-add:
//   msg[e] = cat[x[src[e]] (DIN), edge_attr[e] (4)] @ mw (K x 64) + mb
//   agg[dst[e]] += msg[e]
// One wave = 16 edges x 64 cols; 8 waves / block -> 128 edges / block.
// Weights staged in LDS pair-interleaved: sW2[(k/2)*64 + n] = {w[k][n], w[k+1][n]}
// so each B fragment (k even) is ONE aligned ds_load_b64.
// ---------------------------------------------------------------------------
template <int DIN>
__global__ __launch_bounds__(256) void edge_msg_kernel(
    const float* __restrict__ xin, const int* __restrict__ src,
    const int* __restrict__ dst, const float* __restrict__ eattr,
    const float* __restrict__ mw, const float* __restrict__ mb,
    float* __restrict__ agg) {
  constexpr int K  = DIN + F_E;
  constexpr int KP = (K + 3) & ~3;  // pad K to multiple of 4
  __shared__ v2f sW2[(KP / 2) * Hh];
  float* sWf = (float*)sW2;
  for (int i = threadIdx.x; i < KP * Hh; i += 256) {
    const int k = i >> 6, n = i & 63;
    sWf[(k >> 1) * 128 + n * 2 + (k & 1)] = (i < K * Hh) ? mw[i] : 0.0f;
  }
  __syncthreads();

  const int lane  = threadIdx.x & 31;
  const int wv    = threadIdx.x >> 5;
  const int li    = lane & 15;
  const int hh    = lane >> 4;
  const int ebase = (blockIdx.x * 8 + wv) * 16;  // E = 6250*128 exactly
  const int erow  = ebase + li;
  const int s     = src[erow];

  v8f acc[4] = {};

#pragma unroll
  for (int k = 0; k < KP; k += 4) {
    const int ia = k + 2 * hh;  // always even
    const v2f a = edge_a2<DIN>(xin, eattr, s, erow, k, hh);
#pragma unroll
    for (int n = 0; n < 4; ++n) {
      const v2f b = sW2[(ia >> 1) * Hh + n * 16 + li];
      acc[n] = wmma_f32(a, b, acc[n]);
    }
  }

#pragma unroll
  for (int n = 0; n < 4; ++n) {
    const int col  = n * 16 + li;
    const float bias = mb[col];
#pragma unroll
    for (int r = 0; r < 8; ++r) {
      const int d = dst[ebase + r + 8 * hh];
      atomicAdd(&agg[d * Hh + col], acc[n][r] + bias);
    }
  }
}

// ---------------------------------------------------------------------------
// Node update: h_out = relu(cat[h_in (DIN), agg (64)] @ uw + ub)
// One wave = 16 nodes x 64 cols; tail blocks clamp loads / guard stores.
// ---------------------------------------------------------------------------
template <int DIN>
__global__ __launch_bounds__(256) void update_kernel(
    const float* __restrict__ hin, const float* __restrict__ agg,
    const float* __restrict__ uw, const float* __restrict__ ub,
    float* __restrict__ hout) {
  constexpr int K  = DIN + Hh;      // 75 or 128
  constexpr int KP = (K + 3) & ~3;  // 76 or 128
  __shared__ v2f sW2[(KP / 2) * Hh];  // <= 32 KB
  float* sWf = (float*)sW2;
  for (int i = threadIdx.x; i < KP * Hh; i += 256) {
    const int k = i >> 6, n = i & 63;
    sWf[(k >> 1) * 128 + n * 2 + (k & 1)] = (i < K * Hh) ? uw[i] : 0.0f;
  }
  __syncthreads();

  const int lane  = threadIdx.x & 31;
  const int wv    = threadIdx.x >> 5;
  const int li    = lane & 15;
  const int hh    = lane >> 4;
  const int nbase = (blockIdx.x * 8 + wv) * 16;
  int nrow = nbase + li;
  if (nrow >= Nn) nrow = Nn - 1;  // clamp loads; stores guarded below

  v8f acc[4] = {};

#pragma unroll
  for (int k = 0; k < KP; k += 4) {
    const int ia = k + 2 * hh;  // always even
    const v2f a = upd_a2<DIN>(hin, agg, nrow, k, hh);
#pragma unroll
    for (int n = 0; n < 4; ++n) {
      const v2f b = sW2[(ia >> 1) * Hh + n * 16 + li];
      acc[n] = wmma_f32(a, b, acc[n]);
    }
  }

#pragma unroll
  for (int n = 0; n < 4; ++n) {
    const int col  = n * 16 + li;
    const float bias = ub[col];
#pragma unroll
    for (int r = 0; r < 8; ++r) {
      const int m = nbase + r + 8 * hh;
      if (m < Nn) hout[m * Hh + col] = fmaxf(acc[n][r] + bias, 0.0f);
    }
  }
}

// ---------------------------------------------------------------------------
// Graph mean-pool (phase 1): per-graph sums + counts via L2 atomics
// ---------------------------------------------------------------------------
__global__ void pool_kernel(const float* __restrict__ h,
                            const int* __restrict__ batch,
                            float* __restrict__ gsum, float* __restrict__ gcnt) {
  int i = blockIdx.x * blockDim.x + threadIdx.x;
  if (i >= Nn * Hh) return;
  int n = i >> 6;
  int c = i & 63;
  int g = batch[n];
  atomicAdd(&gsum[g * Hh + c], h[i]);
  if (c == 0) atomicAdd(&gcnt[g], 1.0f);
}

// ---------------------------------------------------------------------------
// Output head: out[g,o] = (gsum[g]/max(cnt,1)) @ out_w + out_b   (tiny GEMM)
// ---------------------------------------------------------------------------
__global__ void out_kernel(const float* __restrict__ gsum,
                           const float* __restrict__ gcnt,
                           const float* __restrict__ ow,
                           const float* __restrict__ ob,
                           float* __restrict__ out) {
  int i = blockIdx.x * blockDim.x + threadIdx.x;
  if (i >= Gg * OUTd) return;
  int g = i / OUTd;
  int o = i % OUTd;
  float inv = 1.0f / fmaxf(gcnt[g], 1.0f);
  float acc = ob[o];
#pragma unroll 8
  for (int c = 0; c < Hh; ++c) acc += gsum[g * Hh + c] * inv * ow[c * OUTd + o];
  out[i] = acc;
}

// ---------------------------------------------------------------------------
// Launcher
// ---------------------------------------------------------------------------
extern "C" void kernel_launch(void* const* d_in, const int* in_sizes, int n_in,
                              void* d_out, int out_size, void* d_ws, size_t ws_size,
                              hipStream_t stream) {
  const float* x   = (const float*)d_in[0];
  const int* ei    = (const int*)d_in[1];   // [2, E]
  const int* batch = (const int*)d_in[2];   // [N]
  const float* ea  = (const float*)d_in[3]; // [E, 4]
  const int* src = ei;
  const int* dst = ei + Ee;

  const float* mw0 = (const float*)d_in[4];
  const float* mb0 = (const float*)d_in[5];
  const float* uw0 = (const float*)d_in[6];
  const float* ub0 = (const float*)d_in[7];
  const float* mw1 = (const float*)d_in[8];
  const float* mb1 = (const float*)d_in[9];
  const float* uw1 = (const float*)d_in[10];
  const float* ub1 = (const float*)d_in[11];
  const float* mw2 = (const float*)d_in[12];
  const float* mb2 = (const float*)d_in[13];
  const float* uw2 = (const float*)d_in[14];
  const float* ub2 = (const float*)d_in[15];
  const float* ow  = (const float*)d_in[16];
  const float* ob  = (const float*)d_in[17];

  float* ws  = (float*)d_ws;
  float* hA  = ws;                 // [N, 64]
  float* hB  = ws + (size_t)Nn * Hh;
  float* agg = ws + (size_t)2 * Nn * Hh;
  float* gsum = agg;               // overlay: agg dead after layer 2 update
  float* gcnt = agg + (size_t)Gg * Hh;

  const int ZB = 256;
  const int nzAgg  = (Nn * Hh + ZB - 1) / ZB;
  const int nzPool = (Gg * Hh + Gg + ZB - 1) / ZB;
  const int edgeBlocks = Ee / 128;             // 6250 (exact)
  const int nodeBlocks = (Nn + 127) / 128;     // 391

  // ---- Layer 0 (din = 11) ----
  zero_kernel<<<nzAgg, ZB, 0, stream>>>(agg, Nn * Hh);
  edge_msg_kernel<11><<<edgeBlocks, 256, 0, stream>>>(x, src, dst, ea, mw0, mb0, agg);
  update_kernel<11><<<nodeBlocks, 256, 0, stream>>>(x, agg, uw0, ub0, hA);

  // ---- Layer 1 (din = 64) ----
  zero_kernel<<<nzAgg, ZB, 0, stream>>>(agg, Nn * Hh);
  edge_msg_kernel<64><<<edgeBlocks, 256, 0, stream>>>(hA, src, dst, ea, mw1, mb1, agg);
  update_kernel<64><<<nodeBlocks, 256, 0, stream>>>(hA, agg, uw1, ub1, hB);

  // ---- Layer 2 (din = 64) ----
  zero_kernel<<<nzAgg, ZB, 0, stream>>>(agg, Nn * Hh);
  edge_msg_kernel<64><<<edgeBlocks, 256, 0, stream>>>(hB, src, dst, ea, mw2, mb2, agg);
  update_kernel<64><<<nodeBlocks, 256, 0, stream>>>(hB, agg, uw2, ub2, hA);

  // ---- Global mean pool + output head ----
  zero_kernel<<<nzPool, ZB, 0, stream>>>(gsum, Gg * Hh + Gg);
  pool_kernel<<<(Nn * Hh + 255) / 256, 256, 0, stream>>>(hA, batch, gsum, gcnt);
  out_kernel<<<(Gg * OUTd + 255) / 256, 256, 0, stream>>>(gsum, gcnt, ow, ob, (float*)d_out);
}